// Sage_32238024524264
// MI455X (gfx1250) — compile-verified
//
#include <hip/hip_runtime.h>

typedef __attribute__((ext_vector_type(2))) float v2f;
typedef __attribute__((ext_vector_type(8))) float v8f;

#define N_NODES 100000
#define F_IN    128
#define HID     256
#define C_OUT   50
#define MTILES  (N_NODES / 16)   // 6250, exact

// ---------------------------------------------------------------------------
// Zero a float buffer (float4-vectorized; n must be a multiple of 4).
// ---------------------------------------------------------------------------
__global__ void zero_f4_kernel(float4* __restrict__ p, long n4) {
    long i = (long)blockIdx.x * blockDim.x + threadIdx.x;
    if (i < n4) p[i] = make_float4(0.f, 0.f, 0.f, 0.f);
}

// ---------------------------------------------------------------------------
// Scatter-add, 128 features per edge. One wave per edge; the 32 lanes cover
// the 128 contiguous floats (float4 per lane) -> 512B coalesced gather and
// coalesced fp32 atomics into agg[dst].
// ---------------------------------------------------------------------------
__global__ void scatter_add_128(const float* __restrict__ x,
                                const int*   __restrict__ src,
                                const int*   __restrict__ dst,
                                float*       __restrict__ agg,
                                int E) {
    int wave = (int)((blockIdx.x * blockDim.x + threadIdx.x) >> 5);
    int lane = threadIdx.x & 31;
    if (wave >= E) return;
    int s = src[wave];
    int d = dst[wave];
    const float4 v = *(const float4*)(x + (long)s * F_IN + lane * 4);
    float* o = agg + (long)d * F_IN + lane * 4;
    unsafeAtomicAdd(o + 0, v.x);
    unsafeAtomicAdd(o + 1, v.y);
    unsafeAtomicAdd(o + 2, v.z);
    unsafeAtomicAdd(o + 3, v.w);
}

// Same, 256 features (h): two float4 chunks per lane.
__global__ void scatter_add_256(const float* __restrict__ h,
                                const int*   __restrict__ src,
                                const int*   __restrict__ dst,
                                float*       __restrict__ agg,
                                int E) {
    int wave = (int)((blockIdx.x * blockDim.x + threadIdx.x) >> 5);
    int lane = threadIdx.x & 31;
    if (wave >= E) return;
    int s = src[wave];
    int d = dst[wave];
    const float* xr = h + (long)s * HID;
    float*       o  = agg + (long)d * HID;
    const float4 v0 = *(const float4*)(xr + lane * 4);
    const float4 v1 = *(const float4*)(xr + 128 + lane * 4);
    unsafeAtomicAdd(o + lane * 4 + 0, v0.x);
    unsafeAtomicAdd(o + lane * 4 + 1, v0.y);
    unsafeAtomicAdd(o + lane * 4 + 2, v0.z);
    unsafeAtomicAdd(o + lane * 4 + 3, v0.w);
    unsafeAtomicAdd(o + 128 + lane * 4 + 0, v1.x);
    unsafeAtomicAdd(o + 128 + lane * 4 + 1, v1.y);
    unsafeAtomicAdd(o + 128 + lane * 4 + 2, v1.z);
    unsafeAtomicAdd(o + 128 + lane * 4 + 3, v1.w);
}

// ---------------------------------------------------------------------------
// Fused SAGE GEMM:  Out = act( Agg*Wl^T + bias + Xin*Wr^T )
//   Agg, Xin : [N_NODES, K]   Wl, Wr : [NOUT, K]   bias : [NOUT]
// One wave computes one 16x16 output tile with V_WMMA_F32_16X16X4_F32,
// chaining 2 WMMAs (agg path + self path) per K=4 step into one accumulator.
//
// f32 16x16x4 operand layout (wave32):
//   A (16x4): lane l holds row l%16, K = (l/16)*2 + v  -> contiguous float2
//   B (4x16): lane l holds col l%16, K = (l/16)*2 + v  -> contiguous float2
//             (B = W^T, so B[k][n] = W[n][k]: float2 load from W row n)
//   C/D (16x16): VGPR j -> M = j + (l/16)*8, N = l%16
// ---------------------------------------------------------------------------
template <int K, int NOUT, bool RELU>
__global__ void sage_gemm(const float* __restrict__ Agg,
                          const float* __restrict__ Xin,
                          const float* __restrict__ Wl,
                          const float* __restrict__ Wr,
                          const float* __restrict__ bias,
                          float*       __restrict__ Out,
                          int ntilesN) {
    const int waveId = (int)((blockIdx.x * blockDim.x + threadIdx.x) >> 5);
    const int lane   = threadIdx.x & 31;
    const int mtile  = waveId / ntilesN;
    const int ntile  = waveId % ntilesN;
    if (mtile >= MTILES) return;           // wave-uniform; EXEC stays all-ones

    const int half16 = lane & 15;          // A row / B column within tile
    const int kh     = (lane >> 4) << 1;   // K sub-pair: 0 or 2
    const int m      = mtile * 16 + half16;              // M is exact (6250*16)
    const int ncol   = ntile * 16 + half16;
    const int ncolB  = (ncol < NOUT) ? ncol : (NOUT - 1);  // clamp for loads

    const float* aptr  = Agg + (long)m * K + kh;
    const float* xptr  = Xin + (long)m * K + kh;
    const float* wlptr = Wl  + (long)ncolB * K + kh;
    const float* wrptr = Wr  + (long)ncolB * K + kh;

    v8f acc = {};
    #pragma unroll 4
    for (int k = 0; k < K; k += 4) {
        v2f a1 = *(const v2f*)(aptr + k);
        v2f b1 = *(const v2f*)(wlptr + k);
        acc = __builtin_amdgcn_wmma_f32_16x16x4_f32(
            false, a1, false, b1, (short)0, acc, false, false);
        v2f a2 = *(const v2f*)(xptr + k);
        v2f b2 = *(const v2f*)(wrptr + k);
        acc = __builtin_amdgcn_wmma_f32_16x16x4_f32(
            false, a2, false, b2, (short)0, acc, false, false);
    }

    const float bv = bias[ncolB];
    const int mbase = mtile * 16 + ((lane >> 4) << 3);   // +8 for upper half
    if (ncol < NOUT) {                                    // post-WMMA predication
        #pragma unroll
        for (int j = 0; j < 8; ++j) {
            float v = acc[j] + bv;
            if (RELU) v = fmaxf(v, 0.f);
            Out[(long)(mbase + j) * NOUT + ncol] = v;
        }
    }
}

// ---------------------------------------------------------------------------
// Driver
// ---------------------------------------------------------------------------
extern "C" void kernel_launch(void* const* d_in, const int* in_sizes, int n_in,
                              void* d_out, int out_size, void* d_ws, size_t ws_size,
                              hipStream_t stream) {
    const float* x   = (const float*)d_in[0];
    const int*   ei  = (const int*)  d_in[1];
    const float* W1l = (const float*)d_in[2];
    const float* b1l = (const float*)d_in[3];
    const float* W1r = (const float*)d_in[4];
    const float* W2l = (const float*)d_in[5];
    const float* b2l = (const float*)d_in[6];
    const float* W2r = (const float*)d_in[7];
    float* out = (float*)d_out;

    const int E = in_sizes[1] / 2;        // edge_index is [2, E]
    const int* src = ei;
    const int* dst = ei + E;

    // Workspace layout: agg1 | agg2 | h  (agg1+agg2 zeroed contiguously)
    float* agg1 = (float*)d_ws;                              // [N, 128]
    float* agg2 = agg1 + (long)N_NODES * F_IN;               // [N, 256]
    float* h    = agg2 + (long)N_NODES * HID;                // [N, 256]

    // 1) zero both aggregation buffers (N * (128+256) floats, contiguous)
    {
        long n4 = (long)N_NODES * (F_IN + HID) / 4;
        int blocks = (int)((n4 + 255) / 256);
        zero_f4_kernel<<<blocks, 256, 0, stream>>>((float4*)agg1, n4);
    }
    // 2) layer-1 neighborhood sum
    scatter_add_128<<<(E + 7) / 8, 256, 0, stream>>>(x, src, dst, agg1, E);
    // 3) h = relu(agg1*W1l^T + b1l + x*W1r^T)     [6250*16 waves]
    {
        int waves = MTILES * (HID / 16);
        sage_gemm<F_IN, HID, true><<<waves / 8, 256, 0, stream>>>(
            agg1, x, W1l, W1r, b1l, h, HID / 16);
    }
    // 4) layer-2 neighborhood sum
    scatter_add_256<<<(E + 7) / 8, 256, 0, stream>>>(h, src, dst, agg2, E);
    // 5) out = agg2*W2l^T + b2l + h*W2r^T          [6250*4 waves, N=50 partial]
    {
        int ntilesN = (C_OUT + 15) / 16;              // 4
        int waves   = MTILES * ntilesN;
        sage_gemm<HID, C_OUT, false><<<waves / 8, 256, 0, stream>>>(
            agg2, h, W2l, W2r, b2l, out, ntilesN);
    }
}